// Stem_slic_71081708748828
// MI455X (gfx1250) — compile-verified
//
#include <hip/hip_runtime.h>
#include <hip/hip_bf16.h>

typedef float v2f __attribute__((ext_vector_type(2)));
typedef float v8f __attribute__((ext_vector_type(8)));

#define IMG_W   224
#define HW      50176           // 224*224
#define SEG     196             // 14*14
#define NBATCH  128
#define M_ROWS  (NBATCH * SEG)  // 25088
#define BN_EPS  1e-5f
#define CHUNKS  8               // pixel chunks per image for seg accumulation

// ---------------------------------------------------------------------------
// Pass-through copy of `sliced` (int32 bits) into the front of d_out.
// ---------------------------------------------------------------------------
__global__ void copy_sliced_kernel(const int4* __restrict__ src,
                                   int4* __restrict__ dst, int n4) {
    int i = blockIdx.x * blockDim.x + threadIdx.x;
    if (i < n4) dst[i] = src[i];
}

__global__ void zero_f32_kernel(float* __restrict__ p, int n) {
    int i = blockIdx.x * blockDim.x + threadIdx.x;
    if (i < n) p[i] = 0.0f;
}

// Pad w0 (96x11) -> (96x16) with zeros so K is a multiple of 4 for WMMA.
__global__ void pad_w0_kernel(const float* __restrict__ w0,
                              float* __restrict__ w0p) {
    int t = blockIdx.x * blockDim.x + threadIdx.x;
    if (t < 96 * 16) {
        int n = t >> 4, k = t & 15;
        w0p[t] = (k < 11) ? w0[n * 11 + k] : 0.0f;
    }
}

// ---------------------------------------------------------------------------
// Stage 1a: per-segment sums. One block = one (image, pixel-chunk).
// LDS accumulators (196 segs x 9 f32 = 7KB) -> one global atomic flush/block.
// acc layout: [b*SEG + s][9] = {cnt, rowSum, colSum, r, g, b, r2, g2, b2}
// ---------------------------------------------------------------------------
__global__ __launch_bounds__(256)
void seg_accum_kernel(const float* __restrict__ x,
                      const int* __restrict__ sliced,
                      float* __restrict__ acc) {
    __shared__ float lacc[SEG * 9];
    const int b     = blockIdx.x / CHUNKS;
    const int chunk = blockIdx.x % CHUNKS;

    for (int i = threadIdx.x; i < SEG * 9; i += blockDim.x) lacc[i] = 0.0f;
    __syncthreads();

    const int pxPerChunk = HW / CHUNKS;        // 6272
    const int p0 = chunk * pxPerChunk;
    const float* xr  = x + (size_t)b * 3 * HW;
    const int*   ids = sliced + (size_t)b * HW;

    for (int p = p0 + threadIdx.x; p < p0 + pxPerChunk; p += blockDim.x) {
        int   id  = ids[p];
        float row = (float)(p / IMG_W);
        float col = (float)(p % IMG_W);
        float r  = xr[p];
        float g  = xr[HW + p];
        float bl = xr[2 * HW + p];
        float* s = lacc + id * 9;
        atomicAdd(s + 0, 1.0f);
        atomicAdd(s + 1, row);
        atomicAdd(s + 2, col);
        atomicAdd(s + 3, r);
        atomicAdd(s + 4, g);
        atomicAdd(s + 5, bl);
        atomicAdd(s + 6, r * r);
        atomicAdd(s + 7, g * g);
        atomicAdd(s + 8, bl * bl);
    }
    __syncthreads();

    float* gacc = acc + (size_t)b * SEG * 9;
    for (int i = threadIdx.x; i < SEG * 9; i += blockDim.x)
        atomicAdd(gacc + i, lacc[i]);
}

// ---------------------------------------------------------------------------
// Stage 1b: finalize 11 features per segment, zero-pad to 16 channels.
// feat layout: [M_ROWS][16] row-major (GEMM A-matrix for layer 0, K=16).
// ---------------------------------------------------------------------------
__global__ void seg_finalize_kernel(const float* __restrict__ acc,
                                    const float* __restrict__ x,
                                    float* __restrict__ feat) {
    int idx = blockIdx.x * blockDim.x + threadIdx.x;
    if (idx >= M_ROWS) return;
    int b = idx / SEG;
    const float* a = acc + (size_t)idx * 9;

    float cnt  = a[0];
    float safe = fmaxf(cnt, 1.0f);
    float inv  = 1.0f / safe;
    float mrow = a[1] * inv, mcol = a[2] * inv;
    float mr = a[3] * inv, mg = a[4] * inv, mb = a[5] * inv;
    float den = 1.0f / fmaxf(cnt - 1.0f, 1.0f);
    float vr = (a[6] - cnt * mr * mr) * den;
    float vg = (a[7] - cnt * mg * mg) * den;
    float vb = (a[8] - cnt * mb * mb) * den;
    bool  many = cnt > 1.0f;
    float sr = many ? sqrtf(fmaxf(vr, 0.0f)) : 0.0f;
    float sg = many ? sqrtf(fmaxf(vg, 0.0f)) : 0.0f;
    float sb = many ? sqrtf(fmaxf(vb, 0.0f)) : 0.0f;

    int ri = min(max((int)mrow, 0), IMG_W - 1);   // trunc == floor (>=0)
    int ci = min(max((int)mcol, 0), IMG_W - 1);
    const float* xr = x + (size_t)b * 3 * HW + ri * IMG_W + ci;
    float g0 = xr[0], g1 = xr[HW], g2 = xr[2 * HW];

    float mask = (cnt > 0.0f) ? 1.0f : 0.0f;
    float* f = feat + (size_t)idx * 16;
    f[0] = mrow * mask;  f[1] = mcol * mask;
    f[2] = mr * mask;    f[3] = mg * mask;    f[4] = mb * mask;
    f[5] = sr * mask;    f[6] = sg * mask;    f[7] = sb * mask;
    f[8] = g0 * mask;    f[9] = g1 * mask;    f[10] = g2 * mask;
    f[11] = 0.0f; f[12] = 0.0f; f[13] = 0.0f; f[14] = 0.0f; f[15] = 0.0f;
}

// ---------------------------------------------------------------------------
// Stage 2: fp32 WMMA GEMM + fused BN (+ReLU), compile-time shapes.
//   Y[m][n] = relu?( (sum_k A[m][k]*W[n][k]) * alpha[n] + beta[n] )
// Wave tile: 32(M) x 16*NF(N): 2 A-frags x NF B-frags, 2*NF WMMAs per k-step.
// K/N are template constants so the i*16*K / j*16*K row offsets fold into
// the 24-bit global_load immediate: all fragment loads hang off two base
// pointers and the scheduler can batch the whole load group per k-step.
// Fragment layout per V_WMMA_F32_16X16X4_F32 (wave32):
//   A/B frag: lane L loads float2 at row (base + L%16), k + 2*(L/16)
//   C/D frag: vgpr r -> row base + 8*(L/16) + r, col base + (L%16)
// ---------------------------------------------------------------------------
template <int NF, int K, int N>
__global__ __launch_bounds__(256)
void gemm_bn_relu_wmma(const float* __restrict__ A,
                       const float* __restrict__ W,
                       const float* __restrict__ bias,
                       const float* __restrict__ gam,
                       const float* __restrict__ bet,
                       const float* __restrict__ rmean,
                       const float* __restrict__ rvar,
                       float* __restrict__ Y,
                       int act) {
    const int lane = threadIdx.x & 31;
    const int wave = threadIdx.x >> 5;
    const int gw   = blockIdx.x * 8 + wave;       // 8 waves / block
    constexpr int nTiles = N / (16 * NF);
    const int mTile = gw / nTiles;
    const int nTile = gw % nTiles;
    const int m0 = mTile << 5;
    const int n0 = nTile * (16 * NF);
    const int mi   = lane & 15;
    const int half = lane >> 4;

    const float* arow = A + (size_t)(m0 + mi) * K + 2 * half;
    const float* brow = W + (size_t)(n0 + mi) * K + 2 * half;

    v8f acc[2][NF];
    #pragma unroll
    for (int i = 0; i < 2; ++i)
        #pragma unroll
        for (int j = 0; j < NF; ++j) acc[i][j] = (v8f){};

    #pragma unroll 4
    for (int k = 0; k < K; k += 4) {
        v2f a0 = *(const v2f*)(arow + k);
        v2f a1 = *(const v2f*)(arow + k + 16 * K);
        v2f b[NF];
        #pragma unroll
        for (int j = 0; j < NF; ++j)
            b[j] = *(const v2f*)(brow + k + j * 16 * K);
        #pragma unroll
        for (int j = 0; j < NF; ++j) {
            acc[0][j] = __builtin_amdgcn_wmma_f32_16x16x4_f32(
                false, a0, false, b[j], (short)0, acc[0][j], false, false);
            acc[1][j] = __builtin_amdgcn_wmma_f32_16x16x4_f32(
                false, a1, false, b[j], (short)0, acc[1][j], false, false);
        }
    }

    // Epilogue: per-column BN params (column is fixed per lane within a frag).
    #pragma unroll
    for (int j = 0; j < NF; ++j) {
        int col = n0 + (j << 4) + mi;
        float al = gam[col] * rsqrtf(rvar[col] + BN_EPS);
        float bt = (bias[col] - rmean[col]) * al + bet[col];
        #pragma unroll
        for (int i = 0; i < 2; ++i) {
            #pragma unroll
            for (int r = 0; r < 8; ++r) {
                int row = m0 + (i << 4) + (half << 3) + r;
                float y = acc[i][j][r] * al + bt;
                if (act) y = fmaxf(y, 0.0f);
                Y[(size_t)row * N + col] = y;
            }
        }
    }
}

// ---------------------------------------------------------------------------
// Final transpose: [b*SEG+s][768] -> (B,768,14,14). Coalesced writes;
// strided reads come out of the 192MB L2 (h5 is 77MB, just written).
// ---------------------------------------------------------------------------
__global__ void transpose_out_kernel(const float* __restrict__ h5,
                                     float* __restrict__ out) {
    int tid = blockIdx.x * blockDim.x + threadIdx.x;
    if (tid >= NBATCH * 768 * SEG) return;
    int s = tid % SEG;
    int o = (tid / SEG) % 768;
    int b = tid / (SEG * 768);
    out[tid] = h5[((size_t)b * SEG + s) * 768 + o];
}

// ---------------------------------------------------------------------------
extern "C" void kernel_launch(void* const* d_in, const int* in_sizes, int n_in,
                              void* d_out, int out_size, void* d_ws, size_t ws_size,
                              hipStream_t stream) {
    const float* x      = (const float*)d_in[0];
    const int*   sliced = (const int*)d_in[1];
    // layer i params at d_in[2 + 6*i .. 2 + 6*i + 5] = {w, b, g, be, rm, rv}

    // Workspace layout (bytes):
    uint8_t* ws = (uint8_t*)d_ws;
    float* acc   = (float*)(ws);                         // 25088*9*4   =   903168
    float* w0p   = (float*)(ws + 903168);                // 96*16*4     =     6144
    float* feat0 = (float*)(ws + 909312);                // 25088*16*4  =  1605632
    float* bufA  = (float*)(ws + 2514944);               // 25088*768*4 = 77070336
    float* bufB  = (float*)(ws + 2514944 + 77070336);    // 77070336
    (void)ws_size; (void)in_sizes; (void)n_in; (void)out_size;

    // 1) pass-through sliced -> d_out front (bit copy of int32)
    {
        int n4 = (NBATCH * HW) / 4;                      // 1605632
        copy_sliced_kernel<<<n4 / 256, 256, 0, stream>>>(
            (const int4*)sliced, (int4*)d_out, n4);
    }
    // 2) zero segment accumulators
    {
        int n = M_ROWS * 9;
        zero_f32_kernel<<<(n + 255) / 256, 256, 0, stream>>>(acc, n);
    }
    // 3) pad w0 to K=16
    pad_w0_kernel<<<6, 256, 0, stream>>>((const float*)d_in[2], w0p);

    // 4) segment accumulation (LDS atomics, flush via global atomics)
    seg_accum_kernel<<<NBATCH * CHUNKS, 256, 0, stream>>>(x, sliced, acc);

    // 5) finalize per-segment features
    seg_finalize_kernel<<<(M_ROWS + 255) / 256, 256, 0, stream>>>(acc, x, feat0);

    // 6) five WMMA GEMM + BN(+ReLU) layers; ping-pong bufA/bufB
    //    blocks = (M/32) * (N / (16*NF)) / 8 waves-per-block, all exact.
    const float* p[5][6];
    for (int i = 0; i < 5; ++i)
        for (int j = 0; j < 6; ++j) p[i][j] = (const float*)d_in[2 + 6 * i + j];

    constexpr int MT = M_ROWS / 32;                      // 784 (divisible by 8)
    gemm_bn_relu_wmma<2, 16, 96><<<MT * 3 / 8, 256, 0, stream>>>(
        feat0, w0p,     p[0][1], p[0][2], p[0][3], p[0][4], p[0][5], bufA, 1);
    gemm_bn_relu_wmma<4, 96, 192><<<MT * 3 / 8, 256, 0, stream>>>(
        bufA, p[1][0],  p[1][1], p[1][2], p[1][3], p[1][4], p[1][5], bufB, 1);
    gemm_bn_relu_wmma<4, 192, 384><<<MT * 6 / 8, 256, 0, stream>>>(
        bufB, p[2][0],  p[2][1], p[2][2], p[2][3], p[2][4], p[2][5], bufA, 1);
    gemm_bn_relu_wmma<4, 384, 768><<<MT * 12 / 8, 256, 0, stream>>>(
        bufA, p[3][0],  p[3][1], p[3][2], p[3][3], p[3][4], p[3][5], bufB, 1);
    gemm_bn_relu_wmma<4, 768, 768><<<MT * 12 / 8, 256, 0, stream>>>(
        bufB, p[4][0],  p[4][1], p[4][2], p[4][3], p[4][4], p[4][5], bufA, 0);

    // 7) transpose h5 -> (B,768,14,14) at d_out tail
    {
        float* outh = (float*)d_out + (size_t)NBATCH * HW;
        int n = NBATCH * 768 * SEG;
        transpose_out_kernel<<<(n + 255) / 256, 256, 0, stream>>>(bufA, outh);
    }
}